// SpatialBorderLoss_68195490726545
// MI455X (gfx1250) — compile-verified
//
#include <hip/hip_runtime.h>
#include <hip/hip_bf16.h>

typedef __attribute__((ext_vector_type(2))) float v2f;
typedef __attribute__((ext_vector_type(8))) float v8f;

// Exact 32-lane wave reduction using V_WMMA_F32_16X16X4_F32 (wave32, CDNA5).
// Stage 1: A = {x, 0} per lane (A[m,0]=x[m], A[m,2]=x[m+16]), B = ones:
//   D[m,n] = x[m] + x[m+16]  -> VGPR r holds pairsum[r] (lanes 0-15) / pairsum[r+8] (lanes 16-31)
// Stage 2: A = ones, B = (d[2i], d[2i+1]), accumulate through C over i=0..3:
//   every element of acc = sum of all 16 pair-sums = full 32-lane sum.
__device__ __forceinline__ float wave_reduce32_wmma(float x) {
  v2f a;    a.x = x;    a.y = 0.0f;
  v2f ones; ones.x = 1.0f; ones.y = 1.0f;
  v8f zero = {};
  v8f d1 = __builtin_amdgcn_wmma_f32_16x16x4_f32(
      false, a, false, ones, (short)0, zero, false, false);
  v8f acc = {};
#pragma unroll
  for (int i = 0; i < 4; ++i) {
    v2f b; b.x = d1[2 * i]; b.y = d1[2 * i + 1];
    acc = __builtin_amdgcn_wmma_f32_16x16x4_f32(
        false, ones, false, b, (short)0, acc, false, false);
  }
  return acc[0];
}

__global__ void sbl_init(double* __restrict__ acc) {
  acc[0] = 0.0; acc[1] = 0.0; acc[2] = 0.0;
}

__global__ void __launch_bounds__(256) sbl_main(
    const float* __restrict__ pts, const float* __restrict__ gtb,
    const float* __restrict__ w, double* __restrict__ acc, int N) {
  const int tid = threadIdx.x;
  const int gsize = gridDim.x * blockDim.x;
  float s = 0.0f, co = 0.0f, cw = 0.0f;

  for (int n = blockIdx.x * blockDim.x + tid; n < N; n += gsize) {
    // bbox row: 32 bytes, 16B aligned -> two b128 loads
    const float4* bb = (const float4*)(gtb + (size_t)n * 8);
    float4 b0 = bb[0];
    float4 b1 = bb[1];
    float vx[4] = {b0.x, b0.z, b1.x, b1.z};
    float vy[4] = {b0.y, b0.w, b1.y, b1.w};

    float vix[4], viy[4], vjy[4], slope[4];
#pragma unroll
    for (int e = 0; e < 4; ++e) {
      const int j = (e + 3) & 3;  // roll(v, 1): edge (v_e, v_{e-1})
      vix[e] = vx[e];
      viy[e] = vy[e];
      vjy[e] = vy[j];
      slope[e] = (vx[j] - vx[e]) / (vy[j] - vy[e]);  // NaN/Inf masked by straddle
    }
    const float cx = (vx[0] + vx[2]) * 0.5f;
    const float cy = (vy[0] + vy[2]) * 0.5f;

    // pts row: 72 bytes, 8B aligned -> nine b64 loads
    const float2* P = (const float2*)(pts + (size_t)n * 18);
#pragma unroll
    for (int p = 0; p < 9; ++p) {
      const float2 pt = P[p];
      const float Px = pt.x, Py = pt.y;
      int cross = 0;
#pragma unroll
      for (int e = 0; e < 4; ++e) {
        const bool st = (viy[e] > Py) != (vjy[e] > Py);
        const float xi = fmaf(slope[e], Py - viy[e], vix[e]);
        cross += (st && (Px < xi)) ? 1 : 0;
      }
      const float dx = Px - cx;
      const float dy = Py - cy;
      const float dist = 0.2f * sqrtf(fmaf(dx, dx, dy * dy));
      const bool outside = ((cross & 1) == 0);
      s  += outside ? dist : 0.0f;
      co += outside ? 1.0f : 0.0f;
    }
    cw += (w[n] > 0.0f) ? 1.0f : 0.0f;
  }

  // Block reduction: LDS to wave 0, then WMMA wave reduction, then f64 atomics.
  __shared__ float sS[256], sCo[256], sCw[256];
  sS[tid] = s; sCo[tid] = co; sCw[tid] = cw;
  __syncthreads();

  if (tid < 32) {  // whole wave 0 active -> EXEC all ones for WMMA
    float a = 0.0f, b = 0.0f, c = 0.0f;
#pragma unroll
    for (int i = 0; i < 8; ++i) {
      a += sS[tid + 32 * i];
      b += sCo[tid + 32 * i];
      c += sCw[tid + 32 * i];
    }
    a = wave_reduce32_wmma(a);
    b = wave_reduce32_wmma(b);
    c = wave_reduce32_wmma(c);
    if (tid == 0) {
      atomicAdd(&acc[0], (double)a);   // global_atomic_add_f64
      atomicAdd(&acc[1], (double)b);
      atomicAdd(&acc[2], (double)c);
    }
  }
}

__global__ void sbl_final(const double* __restrict__ acc, float* __restrict__ out) {
  const float s  = (float)acc[0];
  const float co = (float)acc[1];
  const float cw = (float)acc[2];
  const float loss = (co > 0.0f) ? (s / fmaxf(co, 1.0f)) : 0.0f;
  out[0] = loss / (cw + 1e-6f);  // loss_weight = 1.0
}

extern "C" void kernel_launch(void* const* d_in, const int* in_sizes, int n_in,
                              void* d_out, int out_size, void* d_ws, size_t ws_size,
                              hipStream_t stream) {
  const float* pts = (const float*)d_in[0];
  const float* gtb = (const float*)d_in[1];
  const float* w   = (const float*)d_in[2];
  const int N = in_sizes[2];  // weight has one element per row
  double* acc = (double*)d_ws;
  float* out = (float*)d_out;

  sbl_init<<<1, 1, 0, stream>>>(acc);
  const int blocks = 4096;  // grid-stride; ~2 rows/thread, plenty of waves for 23.3 TB/s
  sbl_main<<<blocks, 256, 0, stream>>>(pts, gtb, w, acc, N);
  sbl_final<<<1, 1, 0, stream>>>(acc, out);
}